// Dynamic_MultiTeacher7_52699248721954
// MI455X (gfx1250) — compile-verified
//
#include <hip/hip_runtime.h>
#include <math.h>

// ---------------------------------------------------------------------------
// Multi-teacher KD loss, MI455X (gfx1250, wave32).
// B=4096 rows, C=1000 classes, 7 teachers + student; mimic = mean(teachers).
// Bandwidth-bound: 131 MB read once from HBM into LDS (async global->LDS,
// ASYNCcnt path). Per-row exp-sum reductions (Z/S per stream, E1/E20) done on
// the matrix unit via V_WMMA_F32_16X16X4_F32 (streams on M rows, B = ones or
// student broadcast); masked lanes feed exact zeros via argument clamping
// (branch-free inner loop). Global max_preds via monotone atomicMax (order-
// independent -> deterministic). Final combine + mean in tiny kernels.
// ---------------------------------------------------------------------------

#define NROWS 4096
#define NCLS  1000

#ifndef __has_builtin
#define __has_builtin(x) 0
#endif

#if __has_builtin(__builtin_amdgcn_global_load_async_to_lds_b128)
#define USE_ASYNC_LDS 1
typedef int v4i_t __attribute__((ext_vector_type(4)));
typedef __attribute__((address_space(1))) v4i_t as1_v4i;   // global
typedef __attribute__((address_space(3))) v4i_t as3_v4i;   // LDS
#else
#define USE_ASYNC_LDS 0
#endif

#if __has_builtin(__builtin_amdgcn_wmma_f32_16x16x4_f32)
#define USE_WMMA 1
typedef float v2f __attribute__((ext_vector_type(2)));
typedef float v8f __attribute__((ext_vector_type(8)));
#else
#define USE_WMMA 0
#endif

__device__ __forceinline__ void async_wait0() {
#if USE_ASYNC_LDS
#if __has_builtin(__builtin_amdgcn_s_wait_asynccnt)
  __builtin_amdgcn_s_wait_asynccnt(0);
#else
  asm volatile("s_wait_asynccnt 0" ::: "memory");
#endif
#endif
}

// monotone mapping: float order -> unsigned order (for deterministic atomicMax)
__device__ __forceinline__ unsigned f2mono(float f) {
  unsigned u = __float_as_uint(f);
  return (u & 0x80000000u) ? ~u : (u | 0x80000000u);
}
__device__ __forceinline__ float mono2f(unsigned u) {
  return (u & 0x80000000u) ? __uint_as_float(u & 0x7FFFFFFFu)
                           : __uint_as_float(~u);
}

__global__ void mt7_init(unsigned* maxbits) { *maxbits = 0u; }

// ---------------------------------------------------------------------------
// Kernel A: one block (256 thr = 8 waves) per row.
//   LDS: streams 0..6 teachers, 7 student, 8 mimic (1000 f32 each),
//   scalar stats at [9000..9039], WMMA partials at [9040..9199].
// ---------------------------------------------------------------------------
__global__ __launch_bounds__(256) void mt7_rows(
    const float* __restrict__ t0, const float* __restrict__ t1,
    const float* __restrict__ t2, const float* __restrict__ t3,
    const float* __restrict__ t4, const float* __restrict__ t5,
    const float* __restrict__ t6, const float* __restrict__ st,
    const int* __restrict__ tgt, float* __restrict__ rec,
    unsigned* __restrict__ maxbits)
{
  __shared__ float lds[9 * NCLS + 40 + 8 * 20 + 64];   // padded: OOB-safe reads
  const int tid  = threadIdx.x;
  const int lane = tid & 31;
  const int w    = tid >> 5;
  const int row  = blockIdx.x;
  const size_t rb = (size_t)row * NCLS;

  // ---- Phase 0: stage 8 rows (32 KB) into LDS, async global->LDS ----
  const float* srcs[8] = {t0, t1, t2, t3, t4, t5, t6, st};
  if (tid < NCLS / 4) {
#pragma unroll
    for (int s = 0; s < 8; ++s) {
      const float* gp = srcs[s] + rb + tid * 4;
      float* lp = &lds[s * NCLS + tid * 4];
#if USE_ASYNC_LDS
      __builtin_amdgcn_global_load_async_to_lds_b128((as1_v4i*)gp, (as3_v4i*)lp, 0, 0);
#else
      const float4 v = *(const float4*)gp;
      *(float4*)lp = v;
#endif
    }
  }
  async_wait0();
  __syncthreads();

  // ---- Phase 1: mimic = (t1 + ... + t7) / 7  (sequential add order) ----
  for (int c = tid; c < NCLS; c += 256) {
    float m = lds[c];
#pragma unroll
    for (int s = 1; s < 7; ++s) m += lds[s * NCLS + c];
    lds[8 * NCLS + c] = m / 7.0f;
  }
  __syncthreads();

  // ---- Phase 2: per-wave top-2 over its stream; wave0 also student max ----
  const int sb = (w < 7) ? w * NCLS : 8 * NCLS;  // wave 7 -> mimic
  {
    float m1 = -__builtin_inff(), m2 = -__builtin_inff();
    for (int c = lane; c < NCLS; c += 32) {
      const float v = lds[sb + c];
      if (v > m1) { m2 = m1; m1 = v; }
      else if (v > m2) { m2 = v; }
    }
#pragma unroll
    for (int m = 16; m > 0; m >>= 1) {
      const float o1 = __shfl_xor(m1, m, 32);
      const float o2 = __shfl_xor(m2, m, 32);
      const float hi = fmaxf(m1, o1);
      const float lo = fminf(m1, o1);
      m2 = fmaxf(lo, fmaxf(m2, o2));
      m1 = hi;
    }
    if (lane == 0) { lds[9000 + 2 * w] = m1; lds[9001 + 2 * w] = m2; }
    if (w == 0) {
      float mx = -__builtin_inff();
      for (int c = lane; c < NCLS; c += 32) mx = fmaxf(mx, lds[7 * NCLS + c]);
#pragma unroll
      for (int m = 16; m > 0; m >>= 1) mx = fmaxf(mx, __shfl_xor(mx, m, 32));
      if (lane == 0) lds[9016] = mx;
    }
  }
  __syncthreads();

  // ---- Phase 3: exp-sum reductions on the matrix unit ----
  //   Streams on M rows: i=0..6 teachers, 7 mimic, 8 E1(student,T=1),
  //   9 E20(student,T=20), 10..15 zero. Wave w covers columns [w*128, ...).
  //   accZ = A x ones  -> rows: Z_0..7, E1(row8), E20(row9)
  //   accS = A x Bs    -> rows: S_k = sum exp * s
  //   Branch-free masking: clamp exp argument to -1e30 (expf -> exactly 0).
#if USE_WMMA
  {
    const int  i    = lane & 15;
    const int  half = lane >> 4;
    const int  cb   = w * 128;                     // 8-byte aligned pairs
    const int  cend = (cb + 128 < NCLS) ? cb + 128 : NCLS;
    const bool act  = (i < 10);
    const int  base = (i < 7) ? i * NCLS : (i == 7 ? 8 * NCLS : 7 * NCLS);
    const float mt    = (i < 8) ? lds[9000 + 2 * i] : lds[9016];
    const float scale = (i == 8) ? 1.0f : 0.05f;
    const float NEG   = -1.0e30f;

    v8f accZ = {0.f, 0.f, 0.f, 0.f, 0.f, 0.f, 0.f, 0.f};
    v8f accS = {0.f, 0.f, 0.f, 0.f, 0.f, 0.f, 0.f, 0.f};
    const v2f ones = {1.f, 1.f};

    for (int it = 0; it < 32; ++it) {
      const int  c0 = cb + it * 4 + half * 2;      // even -> b64-mergeable
      const bool p0 = act && (c0     < cend);
      const bool p1 = act && (c0 + 1 < cend);
      const bool q0 = (c0     < cend);
      const bool q1 = (c0 + 1 < cend);
      // unconditional in-bounds loads; values masked by cndmask selects
      const float tv0 = lds[base + c0];
      const float tv1 = lds[base + c0 + 1];
      const float sv0 = lds[7 * NCLS + c0];
      const float sv1 = lds[7 * NCLS + c0 + 1];
      const float a0 = p0 ? (tv0 - mt) * scale : NEG;
      const float a1 = p1 ? (tv1 - mt) * scale : NEG;
      const float e0 = expf(a0);                   // 0 exactly when masked
      const float e1 = expf(a1);
      const v2f A  = {e0, e1};
      const v2f Bs = {q0 ? sv0 : 0.f, q1 ? sv1 : 0.f};
      accZ = __builtin_amdgcn_wmma_f32_16x16x4_f32(false, A, false, ones,
                                                   (short)0, accZ, false, false);
      accS = __builtin_amdgcn_wmma_f32_16x16x4_f32(false, A, false, Bs,
                                                   (short)0, accS, false, false);
    }
    if (lane == 0) {
#pragma unroll
      for (int r = 0; r < 8; ++r) {
        lds[9040 + w * 20 + r]     = accZ[r];   // Z partials, streams 0..7
        lds[9040 + w * 20 + 8 + r] = accS[r];   // S partials, streams 0..7
      }
    }
    if (lane == 16) {
      lds[9040 + w * 20 + 16] = accZ[0];        // row 8  -> E1 partial
      lds[9040 + w * 20 + 17] = accZ[1];        // row 9  -> E20 partial
    }
  }
#else
  {
    const float mt = lds[9000 + 2 * w];
    float Z = 0.f, S = 0.f;
    for (int c = lane; c < NCLS; c += 32) {
      const float tv = lds[sb + c];
      const float sv = lds[7 * NCLS + c];
      const float e  = expf((tv - mt) * 0.05f);
      Z += e; S += e * sv;
    }
#pragma unroll
    for (int m = 16; m > 0; m >>= 1) {
      Z += __shfl_xor(Z, m, 32);
      S += __shfl_xor(S, m, 32);
    }
    if (lane == 0) { lds[9018 + 2 * w] = Z; lds[9019 + 2 * w] = S; }
    if (w == 0) {
      const float sm = lds[9016];
      float E1 = 0.f, E20 = 0.f;
      for (int c = lane; c < NCLS; c += 32) {
        const float sv = lds[7 * NCLS + c];
        E1  += expf(sv - sm);
        E20 += expf((sv - sm) * 0.05f);
      }
#pragma unroll
      for (int m = 16; m > 0; m >>= 1) {
        E1  += __shfl_xor(E1, m, 32);
        E20 += __shfl_xor(E20, m, 32);
      }
      if (lane == 0) { lds[9034] = E1; lds[9035] = E20; }
    }
  }
#endif
  __syncthreads();

  // ---- Phase 4: per-row scalars (thread 0) ----
  if (tid == 0) {
    float Zk[8], Sk[8], E1, E20;
#if USE_WMMA
    E1 = 0.f; E20 = 0.f;
#pragma unroll
    for (int k = 0; k < 8; ++k) {
      float z = 0.f, s = 0.f;
#pragma unroll
      for (int w2 = 0; w2 < 8; ++w2) {          // fixed order -> deterministic
        z += lds[9040 + w2 * 20 + k];
        s += lds[9040 + w2 * 20 + 8 + k];
      }
      Zk[k] = z; Sk[k] = s;
    }
#pragma unroll
    for (int w2 = 0; w2 < 8; ++w2) {
      E1  += lds[9040 + w2 * 20 + 16];
      E20 += lds[9040 + w2 * 20 + 17];
    }
#else
#pragma unroll
    for (int k = 0; k < 8; ++k) { Zk[k] = lds[9018 + 2 * k]; Sk[k] = lds[9019 + 2 * k]; }
    E1 = lds[9034]; E20 = lds[9035];
#endif

    const int   y  = tgt[row];
    const float sy = lds[7 * NCLS + y];
    const float sm = lds[9016];
    const float CE    = -(sy - sm - logf(E1));           // -log_softmax(s)[y]
    const float lse20 = sm * 0.05f + logf(E20);          // log sum exp(s/20)

    float kd[8], tv[8], gap[8];
    float tmax = -__builtin_inff();
#pragma unroll
    for (int k = 0; k < 8; ++k) {
      const float m1 = lds[9000 + 2 * k];
      const float m2 = lds[9001 + 2 * k];
      tv[k]  = lds[(k < 7 ? k * NCLS : 8 * NCLS) + y];
      // KD = -(sum p * s/20 - lse20) * 400 ; sum p*s/20 = S/(20 Z)
      kd[k]  = -(Sk[k] / (Zk[k] * 20.f) - lse20) * 400.f;
      gap[k] = (tv[k] == m1) ? (m1 - m2) : 0.f;          // value-equality, per ref
      if (k < 7) tmax = fmaxf(tmax, m1);
    }
    // softmax(gap / 2) over the 8 streams
    float gm = gap[0];
#pragma unroll
    for (int k = 1; k < 8; ++k) gm = fmaxf(gm, gap[k]);
    float eg[8], se = 0.f;
#pragma unroll
    for (int k = 0; k < 8; ++k) { eg[k] = expf((gap[k] - gm) * 0.5f); se += eg[k]; }

    float* r = rec + (size_t)row * 32;
    r[0] = CE;
#pragma unroll
    for (int k = 0; k < 8; ++k) {
      r[1 + k]  = kd[k];
      r[9 + k]  = tv[k];
      r[17 + k] = eg[k] / se;
    }
    atomicMax(maxbits, f2mono(tmax));   // order-independent -> deterministic
  }
}

// ---------------------------------------------------------------------------
// Kernel B: 32 blocks x 128 threads; one row per thread, fixed-order block
// tree reduction -> 32 partial sums (deterministic).
// ---------------------------------------------------------------------------
__global__ __launch_bounds__(128) void mt7_combine(
    const float* __restrict__ rec, const unsigned* __restrict__ maxbits,
    float* __restrict__ partial)
{
  __shared__ float red[128];
  const int tid = threadIdx.x;
  const int row = blockIdx.x * 128 + tid;
  const float maxp = mono2f(*maxbits);

  const float* r = rec + (size_t)row * 32;
  const float CE = r[0];
  float loss = 0.f;
#pragma unroll
  for (int k = 0; k < 8; ++k) {
    const float kd = r[1 + k];
    const float wv = 0.8f * r[9 + k] / maxp;
    const float th = r[17 + k];
    loss += th * ((1.f - wv) * CE + wv * kd);
  }
  red[tid] = loss;
  __syncthreads();
#pragma unroll
  for (int s = 64; s > 0; s >>= 1) {
    if (tid < s) red[tid] += red[tid + s];
    __syncthreads();
  }
  if (tid == 0) partial[blockIdx.x] = red[0];
}

// ---------------------------------------------------------------------------
// Kernel C: single wave32 reduces 32 partials (fixed xor tree), writes mean.
// ---------------------------------------------------------------------------
__global__ __launch_bounds__(32) void mt7_final(
    const float* __restrict__ partial, float* __restrict__ out)
{
  const int lane = threadIdx.x;
  float v = partial[lane];
#pragma unroll
  for (int m = 16; m > 0; m >>= 1) v += __shfl_xor(v, m, 32);
  if (lane == 0) out[0] = v / (float)NROWS;
}

// ---------------------------------------------------------------------------
extern "C" void kernel_launch(void* const* d_in, const int* in_sizes, int n_in,
                              void* d_out, int out_size, void* d_ws, size_t ws_size,
                              hipStream_t stream) {
  (void)in_sizes; (void)n_in; (void)out_size; (void)ws_size;
  const float* t0 = (const float*)d_in[0];
  const float* t1 = (const float*)d_in[1];
  const float* t2 = (const float*)d_in[2];
  const float* t3 = (const float*)d_in[3];
  const float* t4 = (const float*)d_in[4];
  const float* t5 = (const float*)d_in[5];
  const float* t6 = (const float*)d_in[6];
  const float* st = (const float*)d_in[7];
  const int*  tgt = (const int*)d_in[8];
  float* out = (float*)d_out;

  unsigned* maxbits = (unsigned*)d_ws;                       // 4 B
  float* rec     = (float*)((char*)d_ws + 256);              // 4096*32 f32
  float* partial = rec + (size_t)NROWS * 32;                 // 32 f32

  mt7_init<<<1, 1, 0, stream>>>(maxbits);
  mt7_rows<<<NROWS, 256, 0, stream>>>(t0, t1, t2, t3, t4, t5, t6, st, tgt,
                                      rec, maxbits);
  mt7_combine<<<NROWS / 128, 128, 0, stream>>>(rec, maxbits, partial);
  mt7_final<<<1, 32, 0, stream>>>(partial, out);
}